// GaussianMultiheadAttention_22617297780909
// MI455X (gfx1250) — compile-verified
//
#include <hip/hip_runtime.h>
#include <hip/hip_bf16.h>
#include <math.h>

// ---------------------------------------------------------------------------
// Gaussian Multihead Attention, fp32, CDNA5 (gfx1250) WMMA implementation.
//   E=1024 H=16 D=64 T=S=1024 B=2
// GEMMs: V_WMMA_F32_16X16X4_F32, 128x64 block tile, 4 wave32 per block,
// double-buffered LDS staging via GLOBAL_LOAD_ASYNC_TO_LDS_B128 when the
// toolchain exposes it (ASYNCcnt path), float4 sync copies otherwise.
// ---------------------------------------------------------------------------

typedef __attribute__((ext_vector_type(2))) float v2f;
typedef __attribute__((ext_vector_type(8))) float v8f;
typedef __attribute__((ext_vector_type(4))) int   v4i;

#define E_DIM   1024
#define H_HEADS 16
#define D_HEAD  64
#define T_LEN   1024
#define S_LEN   1024
#define B_BATCH 2
#define BH      (B_BATCH * H_HEADS)   // 32
#define ROWS    (T_LEN * B_BATCH)     // 2048
#define SCALE_F 0.125f                // D^-0.5
#define DELTA_F 1.0f

#define TM 128
#define TN 64
#define KC 16
#define LDK 20                        // padded row stride (80B: 16B-aligned, bank-safe)
#define A_BUF (TM * LDK)              // 2560 floats per A buffer
#define B_BUF 1280                    // max(64*20, 16*64) floats per B buffer

#if __has_builtin(__builtin_amdgcn_global_load_async_to_lds_b128) && \
    __has_builtin(__builtin_amdgcn_s_wait_asynccnt)
#define USE_ASYNC_LDS 1
#else
#define USE_ASYNC_LDS 0
#endif

__device__ __forceinline__ void cp16_g2lds(const float* g, float* l)
{
#if USE_ASYNC_LDS
    __builtin_amdgcn_global_load_async_to_lds_b128(
        (__attribute__((address_space(1))) v4i*)g,
        (__attribute__((address_space(3))) v4i*)l, 0, 0);
#else
    *reinterpret_cast<float4*>(l) = *reinterpret_cast<const float4*>(g);
#endif
}

// ---------------------------------------------------------------------------
// Tiled WMMA GEMM.
//   B_IS_KxN == false :  C[m,n] = scale*( sum_k A[m,k]*B[n,k] + bias[n] )   (A·B^T)
//   B_IS_KxN == true  :  C[m,n] = scale*( sum_k A[m,k]*B[k,n] + bias[n] )   (A·B)
// A element (m,k) at A + z*sAz + m*lda + k           (k contiguous)
// B element        at B + z*sBz + n*ldb + k   (NT)   (k contiguous)
//                  at B + z*sBz + k*ldb + n   (NN)   (n contiguous)
// C element (m,n) at C + z*sCz + m*ldc + n
// M multiple of 128, N multiple of 64, K multiple of 16. act==1 -> tanh.
// Block: 128 threads = 4 wave32; wave w owns rows [32w,32w+32): 2x4 subtiles.
// ---------------------------------------------------------------------------
template <bool B_IS_KxN>
__global__ __launch_bounds__(128) void wmma_gemm(
    const float* __restrict__ A, long sAz, int lda,
    const float* __restrict__ B, long sBz, int ldb,
    float* __restrict__ C, long sCz, int ldc,
    const float* __restrict__ bias,
    int M, int N, int K, float scale, int act)
{
    __shared__ float As[2 * A_BUF];
    __shared__ float Bs[2 * B_BUF];

    const int tid  = threadIdx.x;
    const int wave = tid >> 5;
    const int lane = tid & 31;
    const int half = lane >> 4;   // 0: lanes 0-15, 1: lanes 16-31
    const int l16  = lane & 15;

    const int z  = blockIdx.z;
    const int m0 = blockIdx.y * TM;
    const int n0 = blockIdx.x * TN;

    const float* Ab = A + (long)z * sAz;
    const float* Bb = B + (long)z * sBz;
    float*       Cb = C + (long)z * sCz;

    const v8f vzero = {0.f, 0.f, 0.f, 0.f, 0.f, 0.f, 0.f, 0.f};
    v8f acc0[4], acc1[4];
#pragma unroll
    for (int n = 0; n < 4; ++n) { acc0[n] = vzero; acc1[n] = vzero; }

    // ---- stage one KC-deep tile pair into buffer `buf` (6 copies/thread)
    auto stage = [&](int buf, int k0) {
        // A tile: TM x KC, 512 float4 slots across 128 threads
#pragma unroll
        for (int i = 0; i < 4; ++i) {
            int idx = tid + i * 128;
            int m   = idx >> 2;
            int k4  = (idx & 3) * 4;
            const float* g = Ab + (long)(m0 + m) * lda + (k0 + k4);
            cp16_g2lds(g, &As[buf * A_BUF + m * LDK + k4]);
        }
        if (!B_IS_KxN) {
            // B tile: TN x KC (k contiguous), 256 float4 slots
#pragma unroll
            for (int i = 0; i < 2; ++i) {
                int idx = tid + i * 128;
                int n   = idx >> 2;
                int k4  = (idx & 3) * 4;
                const float* g = Bb + (long)(n0 + n) * ldb + (k0 + k4);
                cp16_g2lds(g, &Bs[buf * B_BUF + n * LDK + k4]);
            }
        } else {
            // B tile: KC x TN (n contiguous), 256 float4 slots
#pragma unroll
            for (int i = 0; i < 2; ++i) {
                int idx = tid + i * 128;
                int k   = idx >> 4;
                int n4  = (idx & 15) * 4;
                const float* g = Bb + (long)(k0 + k) * ldb + (n0 + n4);
                cp16_g2lds(g, &Bs[buf * B_BUF + k * TN + n4]);
            }
        }
    };

    // ---- 4 K-steps (K=4 each) x 8 WMMAs on buffer `buf`
    auto compute = [&](int buf) {
        const float* Asb = &As[buf * A_BUF];
        const float* Bsb = &Bs[buf * B_BUF];
#pragma unroll
        for (int kk = 0; kk < KC; kk += 4) {
            const int kb = kk + 2 * half;     // A/B frag: K = 2*half + j
            v2f a0 = *reinterpret_cast<const v2f*>(&Asb[(wave * 32 + l16) * LDK + kb]);
            v2f a1 = *reinterpret_cast<const v2f*>(&Asb[(wave * 32 + 16 + l16) * LDK + kb]);
#pragma unroll
            for (int n = 0; n < 4; ++n) {
                v2f b;
                if (!B_IS_KxN) {
                    b = *reinterpret_cast<const v2f*>(&Bsb[(n * 16 + l16) * LDK + kb]);
                } else {
                    b.x = Bsb[(kb + 0) * TN + n * 16 + l16];
                    b.y = Bsb[(kb + 1) * TN + n * 16 + l16];
                }
                acc0[n] = __builtin_amdgcn_wmma_f32_16x16x4_f32(
                    false, a0, false, b, (short)0, acc0[n], false, false);
                acc1[n] = __builtin_amdgcn_wmma_f32_16x16x4_f32(
                    false, a1, false, b, (short)0, acc1[n], false, false);
            }
        }
    };

    const int nstages = K / KC;
    stage(0, 0);
    for (int s = 0; s < nstages; ++s) {
        const int buf = s & 1;
        if (s + 1 < nstages) stage(buf ^ 1, (s + 1) * KC);
#if USE_ASYNC_LDS
        if (s + 1 < nstages) __builtin_amdgcn_s_wait_asynccnt(6);
        else                 __builtin_amdgcn_s_wait_asynccnt(0);
#endif
        __syncthreads();          // current buffer fully resident in LDS
        compute(buf);
        __syncthreads();          // all reads done before buffer is re-staged
    }

    // ---- epilogue: C VGPR v holds M = v + 8*half, N = l16
#pragma unroll
    for (int n = 0; n < 4; ++n) {
        const int ncol = n0 + n * 16 + l16;
        const float bv = bias ? bias[ncol] : 0.f;
#pragma unroll
        for (int v = 0; v < 8; ++v) {
            const int r0 = m0 + wave * 32 + half * 8 + v;
            float x0 = (acc0[n][v] + bv) * scale;
            float x1 = (acc1[n][v] + bv) * scale;
            if (act == 1) { x0 = tanhf(x0); x1 = tanhf(x1); }
            Cb[(long)r0 * ldc + ncol]        = x0;
            Cb[(long)(r0 + 16) * ldc + ncol] = x1;
        }
    }
}

// ---------------------------------------------------------------------------
// sigma[r] = 1 + DELTA*sigmoid( dot(tanh_wp[r,:], Wvp) + bvp ), r = t*B + b
// ---------------------------------------------------------------------------
__global__ __launch_bounds__(256) void sigma_kernel(
    const float* __restrict__ tmp, const float* __restrict__ Wvp,
    const float* __restrict__ bvp, float* __restrict__ sigma)
{
    __shared__ float red[256];
    const int r = blockIdx.x;
    float acc = 0.f;
    for (int k = threadIdx.x; k < E_DIM; k += 256)
        acc += tmp[(long)r * E_DIM + k] * Wvp[k];
    red[threadIdx.x] = acc;
    __syncthreads();
    for (int s = 128; s > 0; s >>= 1) {
        if (threadIdx.x < (unsigned)s) red[threadIdx.x] += red[threadIdx.x + s];
        __syncthreads();
    }
    if (threadIdx.x == 0) {
        float g = red[0] + bvp[0];
        sigma[r] = 1.0f + DELTA_F * (1.0f / (1.0f + __expf(-g)));
    }
}

// ---------------------------------------------------------------------------
// mean[row] = sum_s softmax(align_logits[row,:])[s] * s    (fused, no probs)
// ---------------------------------------------------------------------------
__global__ __launch_bounds__(256) void mean_kernel(
    const float* __restrict__ logits, float* __restrict__ mean)
{
    __shared__ float red[256];
    const long row = blockIdx.x;
    const float* p = logits + row * (long)S_LEN;

    float mx = -1e30f;
    for (int s = threadIdx.x; s < S_LEN; s += 256) mx = fmaxf(mx, p[s]);
    red[threadIdx.x] = mx;
    __syncthreads();
    for (int s = 128; s > 0; s >>= 1) {
        if (threadIdx.x < (unsigned)s)
            red[threadIdx.x] = fmaxf(red[threadIdx.x], red[threadIdx.x + s]);
        __syncthreads();
    }
    mx = red[0];
    __syncthreads();

    float se = 0.f, sw = 0.f;
    for (int s = threadIdx.x; s < S_LEN; s += 256) {
        float e = __expf(p[s] - mx);
        se += e;
        sw += e * (float)s;
    }
    red[threadIdx.x] = se;
    __syncthreads();
    for (int s = 128; s > 0; s >>= 1) {
        if (threadIdx.x < (unsigned)s) red[threadIdx.x] += red[threadIdx.x + s];
        __syncthreads();
    }
    se = red[0];
    __syncthreads();
    red[threadIdx.x] = sw;
    __syncthreads();
    for (int s = 128; s > 0; s >>= 1) {
        if (threadIdx.x < (unsigned)s) red[threadIdx.x] += red[threadIdx.x + s];
        __syncthreads();
    }
    sw = red[0];
    if (threadIdx.x == 0) mean[row] = sw / se;
}

// ---------------------------------------------------------------------------
// In-place: probs[row,s] = softmax_s( attn[row,s] - (s-mean)^2/(2*sigma^2) )
// row = bh*T + t ; sigma indexed (t,b) row-major; bh = b*H + h
// ---------------------------------------------------------------------------
__global__ __launch_bounds__(256) void softmax_gauss_kernel(
    float* __restrict__ logits, const float* __restrict__ mean,
    const float* __restrict__ sigma)
{
    __shared__ float red[256];
    const long row = blockIdx.x;
    const int bh = (int)(row / T_LEN);
    const int t  = (int)(row % T_LEN);
    const int b  = bh / H_HEADS;
    const float mu = mean[row];
    const float sg = sigma[t * B_BATCH + b];
    const float inv2s2 = 1.0f / (2.0f * sg * sg);
    float* p = logits + row * (long)S_LEN;

    float mx = -1e30f;
    for (int s = threadIdx.x; s < S_LEN; s += 256) {
        float d = (float)s - mu;
        float x = p[s] - d * d * inv2s2;
        p[s] = x;
        mx = fmaxf(mx, x);
    }
    red[threadIdx.x] = mx;
    __syncthreads();
    for (int s = 128; s > 0; s >>= 1) {
        if (threadIdx.x < (unsigned)s)
            red[threadIdx.x] = fmaxf(red[threadIdx.x], red[threadIdx.x + s]);
        __syncthreads();
    }
    mx = red[0];
    __syncthreads();

    float se = 0.f;
    for (int s = threadIdx.x; s < S_LEN; s += 256) {
        float e = __expf(p[s] - mx);
        p[s] = e;
        se += e;
    }
    red[threadIdx.x] = se;
    __syncthreads();
    for (int s = 128; s > 0; s >>= 1) {
        if (threadIdx.x < (unsigned)s) red[threadIdx.x] += red[threadIdx.x + s];
        __syncthreads();
    }
    const float inv = 1.0f / red[0];
    for (int s = threadIdx.x; s < S_LEN; s += 256) p[s] *= inv;
}

// ---------------------------------------------------------------------------
extern "C" void kernel_launch(void* const* d_in, const int* in_sizes, int n_in,
                              void* d_out, int out_size, void* d_ws, size_t ws_size,
                              hipStream_t stream)
{
    const float* query  = (const float*)d_in[0];
    const float* key_in = (const float*)d_in[1];
    const float* value  = (const float*)d_in[2];
    const float* Wq  = (const float*)d_in[3];   const float* bq  = (const float*)d_in[4];
    const float* Wk  = (const float*)d_in[5];   const float* bk  = (const float*)d_in[6];
    const float* Wv  = (const float*)d_in[7];   const float* bv  = (const float*)d_in[8];
    const float* Whq = (const float*)d_in[9];   const float* bhq = (const float*)d_in[10];
    const float* Whk = (const float*)d_in[11];  const float* bhk = (const float*)d_in[12];
    const float* Wwp = (const float*)d_in[13];  const float* bwp = (const float*)d_in[14];
    const float* Wvp = (const float*)d_in[15];  const float* bvp = (const float*)d_in[16];
    const float* Wo  = (const float*)d_in[17];  const float* bo  = (const float*)d_in[18];
    float* out = (float*)d_out;

    // ---- workspace layout (floats) -------------------------------------
    const long BIG  = (long)BH * T_LEN * S_LEN;   // 33,554,432  (128 MB, reused)
    const long PROJ = (long)ROWS * E_DIM;         //  2,097,152  (8 MB each)
    float* ws    = (float*)d_ws;
    float* big   = ws;               // align logits -> (reused) attn logits/probs
    float* q     = big + BIG;
    float* k     = q + PROJ;
    float* v     = k + PROJ;
    float* hq    = v + PROJ;
    float* hk    = hq + PROJ;
    float* tmp   = hk + PROJ;        // tanh(wp) scratch, later attn_out
    float* meanb = tmp + PROJ;       // BH*T
    float* sigb  = meanb + (long)BH * T_LEN;   // ROWS

    const dim3 blk(128);
    const dim3 gproj(E_DIM / TN, ROWS / TM, 1);

    // 1-5: projections  (SCALE folded into q and hq)
    wmma_gemm<false><<<gproj, blk, 0, stream>>>(query, 0, E_DIM, Wq, 0, E_DIM,
        q, 0, E_DIM, bq, ROWS, E_DIM, E_DIM, SCALE_F, 0);
    wmma_gemm<false><<<gproj, blk, 0, stream>>>(key_in, 0, E_DIM, Wk, 0, E_DIM,
        k, 0, E_DIM, bk, ROWS, E_DIM, E_DIM, 1.0f, 0);
    wmma_gemm<false><<<gproj, blk, 0, stream>>>(value, 0, E_DIM, Wv, 0, E_DIM,
        v, 0, E_DIM, bv, ROWS, E_DIM, E_DIM, 1.0f, 0);
    wmma_gemm<false><<<gproj, blk, 0, stream>>>(query, 0, E_DIM, Whq, 0, E_DIM,
        hq, 0, E_DIM, bhq, ROWS, E_DIM, E_DIM, SCALE_F, 0);
    wmma_gemm<false><<<gproj, blk, 0, stream>>>(key_in, 0, E_DIM, Whk, 0, E_DIM,
        hk, 0, E_DIM, bhk, ROWS, E_DIM, E_DIM, 1.0f, 0);

    // 6: tmp = tanh(query @ Wwp^T + bwp)
    wmma_gemm<false><<<gproj, blk, 0, stream>>>(query, 0, E_DIM, Wwp, 0, E_DIM,
        tmp, 0, E_DIM, bwp, ROWS, E_DIM, E_DIM, 1.0f, 1);
    // 7: sigma
    sigma_kernel<<<dim3(ROWS), dim3(256), 0, stream>>>(tmp, Wvp, bvp, sigb);

    // 8: align logits (per head z: base offset z*D in hq/hk — since E == H*D)
    const dim3 gattn(S_LEN / TN, T_LEN / TM, BH);
    wmma_gemm<false><<<gattn, blk, 0, stream>>>(
        hq, D_HEAD, B_BATCH * E_DIM, hk, D_HEAD, B_BATCH * E_DIM,
        big, (long)T_LEN * S_LEN, S_LEN, nullptr,
        T_LEN, S_LEN, D_HEAD, 1.0f, 0);
    // 9: mean over align rows
    mean_kernel<<<dim3(BH * T_LEN), dim3(256), 0, stream>>>(big, meanb);

    // 10: attn logits overwrite `big`
    wmma_gemm<false><<<gattn, blk, 0, stream>>>(
        q, D_HEAD, B_BATCH * E_DIM, k, D_HEAD, B_BATCH * E_DIM,
        big, (long)T_LEN * S_LEN, S_LEN, nullptr,
        T_LEN, S_LEN, D_HEAD, 1.0f, 0);
    // 11: gaussian bias + softmax in place -> probs
    softmax_gauss_kernel<<<dim3(BH * T_LEN), dim3(256), 0, stream>>>(big, meanb, sigb);

    // 12: attn_out = probs @ vh   (NN GEMM; C written in (T,B,E) flat layout)
    const dim3 gout(D_HEAD / TN, T_LEN / TM, BH);
    wmma_gemm<true><<<gout, blk, 0, stream>>>(
        big, (long)T_LEN * S_LEN, S_LEN, v, D_HEAD, B_BATCH * E_DIM,
        tmp, D_HEAD, B_BATCH * E_DIM, nullptr,
        T_LEN, D_HEAD, S_LEN, 1.0f, 0);

    // 13: final projection -> d_out
    const dim3 gfin(E_DIM / TN, ROWS / TM, 1);
    wmma_gemm<false><<<gfin, blk, 0, stream>>>(tmp, 0, E_DIM, Wo, 0, E_DIM,
        out, 0, E_DIM, bo, ROWS, E_DIM, E_DIM, 1.0f, 0);
}